// QuantumDotProductCircuit_91199335563795
// MI455X (gfx1250) — compile-verified
//
#include <hip/hip_runtime.h>
#include <math.h>

// ---------------------------------------------------------------------------
// QuantumDotProductCircuit for MI455X (gfx1250, wave32)
//
// Algebraic collapse: psi = M(W) * ry(angles) with M batch-independent, so
//   ov(b)  = ryq(b)^T * G * ryk(b),  G = Mq^dagger * Mk (fixed 16x16 complex)
//   out(b) = (1 + Re(ov)^2 + Im(ov)^2) / 2
//
// Kernel 1 (1 wave): build G from W_q/W_k (negligible, runs once).
// Kernel 2: per 16-batch tile (one wave):
//   - lower half-lanes compute query-side ry products, upper half key-side
//     (halves the tanh/sin/cos + global-load work vs. both halves duplicating)
//   - Ryk * Gr^T and Ryk * Gi^T via 4+4 chained V_WMMA_F32_16X16X4_F32
//   - D scattered to LDS, per-lane row dot with ryq (Re on lower half, Im on
//     upper), one ds_swizzle SWAPX16 to pair them, coalesced 64B store.
// Memory-bound: ~2*B*64B DRAM line traffic ≈ 2-3 us at 23.3 TB/s; the rework
// above cuts per-wave VALU/DS so compute stays under the memory roof.
// ---------------------------------------------------------------------------

typedef __attribute__((ext_vector_type(2))) float v2f;
typedef __attribute__((ext_vector_type(8))) float v8f;

#define NQ 4
#define NL 2
#define PI_HALF 1.57079632679489662f

struct c32 { float re, im; };
__device__ __forceinline__ c32 cmul(c32 a, c32 b) {
    return { a.re*b.re - a.im*b.im, a.re*b.im + a.im*b.re };
}
__device__ __forceinline__ c32 cadd(c32 a, c32 b) { return { a.re+b.re, a.im+b.im }; }

// Native V_TANH_F32 if the toolchain exposes it, else 3-op exp-based tanh
// (exact at saturation: exp->inf => 1, exp->0 => -1).
__device__ __forceinline__ float fast_tanhf(float x) {
#if __has_builtin(__builtin_amdgcn_tanhf)
    return __builtin_amdgcn_tanhf(x);
#elif __has_builtin(__builtin_amdgcn_tanh_f32)
    return __builtin_amdgcn_tanh_f32(x);
#else
    const float e = __expf(2.0f * x);
    return 1.0f - 2.0f / (e + 1.0f);
#endif
}

// Exchange value between lane l and lane l^16 (SWAPX16 swizzle pattern).
__device__ __forceinline__ float half_swap(float v) {
#if __has_builtin(__builtin_amdgcn_ds_swizzle)
    // group-of-32 mode: offset[15]=0, xor=0x10<<10, and_mask=0x1f -> 0x401F
    return __int_as_float(__builtin_amdgcn_ds_swizzle(__float_as_int(v), 0x401F));
#else
    return __shfl_xor(v, 16, 32);
#endif
}

// ---------------------------------------------------------------------------
// Kernel 1: build G = Mq^dagger * Mk.  One wave: lanes 0-15 evolve basis
// columns of Mq, lanes 16-31 of Mk.  Then 256 entries of G (8 per lane).
// ---------------------------------------------------------------------------
__global__ void __launch_bounds__(32)
build_g_kernel(const float* __restrict__ Wq, const float* __restrict__ Wk,
               float* __restrict__ gr, float* __restrict__ gi)
{
    __shared__ float2 sM[32][16];           // sM[l][s]: column (l&15) of Mq (l<16) / Mk (l>=16)
    const int l   = threadIdx.x;            // 0..31
    const int col = l & 15;
    const float* W = (l >= 16) ? Wk : Wq;

    c32 st[16];
    #pragma unroll
    for (int s = 0; s < 16; ++s) st[s] = { (s == col) ? 1.f : 0.f, 0.f };

    #pragma unroll
    for (int layer = 0; layer < NL; ++layer) {
        // Rot(phi, theta, omega) on wires 0..3  (wire w -> state bit (3-w))
        #pragma unroll
        for (int w = 0; w < NQ; ++w) {
            const float phi = W[(layer*NQ + w)*3 + 0];
            const float th  = W[(layer*NQ + w)*3 + 1];
            const float om  = W[(layer*NQ + w)*3 + 2];
            float ch, sh;  sincosf(0.5f*th,        &sh, &ch);
            float sp, cp;  sincosf(0.5f*(phi+om),  &sp, &cp);
            float sm, cm;  sincosf(0.5f*(phi-om),  &sm, &cm);
            const c32 u00 = {  cp*ch, -sp*ch };
            const c32 u01 = { -cm*sh, -sm*sh };
            const c32 u10 = {  cm*sh, -sm*sh };
            const c32 u11 = {  cp*ch,  sp*ch };
            const int bit = 1 << (3 - w);
            #pragma unroll
            for (int s0 = 0; s0 < 16; ++s0) {
                if ((s0 & bit) == 0) {
                    const int s1 = s0 | bit;
                    const c32 a = st[s0], b = st[s1];
                    st[s0] = cadd(cmul(u00, a), cmul(u01, b));
                    st[s1] = cadd(cmul(u10, a), cmul(u11, b));
                }
            }
        }
        // CNOT(control=w, target=w+1) for w = 0..2
        #pragma unroll
        for (int w = 0; w < NQ-1; ++w) {
            const int cb = 1 << (3 - w);
            const int tb = 1 << (2 - w);
            #pragma unroll
            for (int s0 = 0; s0 < 16; ++s0) {
                if ((s0 & cb) != 0 && (s0 & tb) == 0) {
                    const c32 t = st[s0]; st[s0] = st[s0 | tb]; st[s0 | tb] = t;
                }
            }
        }
    }

    #pragma unroll
    for (int s = 0; s < 16; ++s) sM[l][s] = make_float2(st[s].re, st[s].im);
    __syncthreads();

    // G[i][j] = sum_s conj(Mq[s][i]) * Mk[s][j];  entry e = i*16+j, 8 per lane
    #pragma unroll
    for (int q = 0; q < 8; ++q) {
        const int e = l*8 + q;
        const int i = e >> 4, j = e & 15;
        float ar = 0.f, ai = 0.f;
        #pragma unroll
        for (int s = 0; s < 16; ++s) {
            const float2 a = sM[i][s];       // Mq[s][i]
            const float2 b = sM[16 + j][s];  // Mk[s][j]
            ar += a.x*b.x + a.y*b.y;         // Re(conj(a)*b)
            ai += a.x*b.y - a.y*b.x;         // Im(conj(a)*b)
        }
        gr[e] = ar;
        gi[e] = ai;
    }
}

// ---------------------------------------------------------------------------
// Kernel 2: streaming pass.  256 threads = 8 waves; each wave owns 16 batch
// elements (one WMMA M-tile).  All LDS rows use stride 20 floats (80B): keeps
// float4 16B alignment and maps 16 rows onto distinct banks (20*ml mod 64 all
// distinct); sPk uses stride 12 (same properties).
// ---------------------------------------------------------------------------
__global__ void __launch_bounds__(256)
qdot_wmma_kernel(const float* __restrict__ query, const float* __restrict__ key,
                 const float* __restrict__ gr, const float* __restrict__ gi,
                 float* __restrict__ out, int B, int stride)
{
    __shared__ float sGr[16*20];        // Gr[n][j] at n*20+j
    __shared__ float sGi[16*20];
    __shared__ float sRyq[8][16*20];    // per-wave ryq rows
    __shared__ float sPk [8][16*12];    // per-wave key-side products ph[4],pl[4]
    __shared__ float sDr [8][16*20];    // per-wave D = Ryk*Gr^T rows
    __shared__ float sDi [8][16*20];

    const int tid  = threadIdx.x;
    const int wv   = tid >> 5;
    const int lane = tid & 31;
    const int half = lane >> 4;         // 0: query side / K0-K1 lanes, 1: key side / K2-K3
    const int ml   = lane & 15;

    // broadcast G into LDS (256 threads, 256 entries per array)
    {
        const int n = tid >> 4, j = tid & 15;
        sGr[n*20 + j] = gr[tid];
        sGi[n*20 + j] = gi[tid];
    }
    __syncthreads();                    // only block-wide barrier needed

    const int base = (blockIdx.x * 8 + wv) * 16;
    const int b  = base + ml;
    const int bc = (b < B) ? b : (B - 1);   // clamp: keep EXEC all-1s for WMMA

    // Lower half handles query, upper half key: one load + 4 tanh + 4 sincos/lane
    const float* src = half ? key : query;
    const float4 x4  = *(const float4*)(src + (size_t)bc * stride);

    float c[4], s[4];
    {
        const float xv[4] = { x4.x, x4.y, x4.z, x4.w };
        #pragma unroll
        for (int i = 0; i < 4; ++i) {
            const float a = fast_tanhf(xv[i]) * PI_HALF;   // |a| < pi/2: native sin/cos safe
            s[i] = __sinf(a);
            c[i] = __cosf(a);
        }
    }
    // ry[j] = ph[j>>2] * pl[j&3]  (bit_i(j) = (j >> (3-i)) & 1)
    const float ph[4] = { c[0]*c[1], c[0]*s[1], s[0]*c[1], s[0]*s[1] };
    const float pl[4] = { c[2]*c[3], c[2]*s[3], s[2]*c[3], s[2]*s[3] };

    if (half == 0) {
        // full ryq row for this batch element (4x float4, conflict-free)
        float4* dst = (float4*)&sRyq[wv][ml*20];
        #pragma unroll
        for (int u = 0; u < 4; ++u)
            dst[u] = make_float4(ph[u]*pl[0], ph[u]*pl[1], ph[u]*pl[2], ph[u]*pl[3]);
    } else {
        // key-side factor products; A-operand values built from these by both halves
        float4* dst = (float4*)&sPk[wv][ml*12];
        dst[0] = make_float4(ph[0], ph[1], ph[2], ph[3]);
        dst[1] = make_float4(pl[0], pl[1], pl[2], pl[3]);
    }
    __builtin_amdgcn_wave_barrier();    // wave-private LDS: in-order DS pipe suffices

    // A operand (ryk tile): lane<16 holds (K=4t, 4t+1), lane>=16 holds (K=4t+2, 4t+3)
    const int   kb = 2*half;
    const float4 kh = *(const float4*)&sPk[wv][ml*12];           // ph[0..3]
    const v2f    kl = *(const v2f*)  &sPk[wv][ml*12 + 4 + kb];   // pl[kb], pl[kb+1]
    const float khv[4] = { kh.x, kh.y, kh.z, kh.w };

    // D[m][i] = sum_j ryk[m][j] * G[i][j]; B[j][n] = G[n][j]
    v8f accr = {};
    v8f acci = {};
    #pragma unroll
    for (int t = 0; t < 4; ++t) {
        v2f a;
        a.x = khv[t] * kl.x;
        a.y = khv[t] * kl.y;
        const int g = ml*20 + 4*t + kb;              // even offset -> 8B-aligned b64 loads
        const v2f br = *(const v2f*)&sGr[g];
        const v2f bi = *(const v2f*)&sGi[g];
        accr = __builtin_amdgcn_wmma_f32_16x16x4_f32(false, a, false, br,
                                                     (short)0, accr, false, false);
        acci = __builtin_amdgcn_wmma_f32_16x16x4_f32(false, a, false, bi,
                                                     (short)0, acci, false, false);
    }

    // Re-partition D through LDS: D VGPR p holds M = p + 8*half, N = ml.
    #pragma unroll
    for (int p = 0; p < 8; ++p) {
        const int m = p + 8*half;
        sDr[wv][m*20 + ml] = accr[p];
        sDi[wv][m*20 + ml] = acci[p];
    }
    __builtin_amdgcn_wave_barrier();

    // Row dot products: lane ml of lower half -> Re(ov[ml]), upper half -> Im(ov[ml])
    const float* Drow = half ? &sDi[wv][ml*20] : &sDr[wv][ml*20];
    const float* Rrow = &sRyq[wv][ml*20];
    float acc = 0.f;
    #pragma unroll
    for (int u = 0; u < 4; ++u) {
        const float4 d = *(const float4*)&Drow[4*u];
        const float4 r = *(const float4*)&Rrow[4*u];
        acc += d.x*r.x + d.y*r.y + d.z*r.z + d.w*r.w;
    }

    const float other = half_swap(acc);  // lower lanes receive Im, upper receive Re
    if (half == 0) {
        const int ob = base + ml;        // lanes 0-15: coalesced 64B store per wave
        if (ob < B) out[ob] = 0.5f * (1.0f + acc*acc + other*other);
    }
}

// ---------------------------------------------------------------------------
extern "C" void kernel_launch(void* const* d_in, const int* in_sizes, int n_in,
                              void* d_out, int out_size, void* d_ws, size_t ws_size,
                              hipStream_t stream)
{
    const float* query = (const float*)d_in[0];
    const float* key   = (const float*)d_in[1];
    const float* Wq    = (const float*)d_in[2];
    const float* Wk    = (const float*)d_in[3];
    float* out = (float*)d_out;

    const int B      = out_size;              // 262144
    const int stride = in_sizes[0] / B;       // 64 floats per row

    float* gr = (float*)d_ws;                 // 256 floats
    float* gi = gr + 256;                     // 256 floats

    build_g_kernel<<<1, 32, 0, stream>>>(Wq, Wk, gr, gi);

    const int blocks = (B + 127) / 128;       // 8 waves * 16 batch/wave per block
    qdot_wmma_kernel<<<blocks, 256, 0, stream>>>(query, key, gr, gi, out, B, stride);
}